// RaytracedOptics_68968584839605
// MI455X (gfx1250) — compile-verified
//
#include <hip/hip_runtime.h>
#include <cstdint>

// Problem constants from the reference
#define B_CNT 8
#define F_CNT 21
#define P_CNT 16384
#define W_CNT 3
#define S_CNT 3
#define GROUPS (B_CNT * F_CNT)            // 168 (b,f) groups
#define NELEM  ((long long)GROUPS * P_CNT * W_CNT)  // 8,257,536 per stack-surface

// Workspace layout (floats):
//   [0 .. GROUPS*12)  : per (b*F+f) group, w*4 + {n, sx, sy, s2}
//   [GROUPS*12]       : raw WMMA sumQ accumulator (needs /16 for column
//                       replication, /3 for the /S in the reference)
#define WS_STATS  (GROUPS * 12)           // 2016
#define WS_SUMQ   WS_STATS
#define WS_FLOATS (WS_STATS + 1)

typedef float v2f __attribute__((ext_vector_type(2)));
typedef float v8f __attribute__((ext_vector_type(8)));

__global__ __launch_bounds__(256) void zero_ws_kernel(float* __restrict__ ws) {
  int i = blockIdx.x * blockDim.x + threadIdx.x;
  if (i < WS_FLOATS) ws[i] = 0.0f;
}

// ---------------------------------------------------------------------------
// Kernel 1: masked spot-moment partial sums per (b,f,w).
// Memory layout of x/y/ok is (B,F,P,W) with W innermost, so within one (b,f)
// group the region is P*W = 49152 contiguous floats with w = idx % 3.
// Each thread handles one 12-float chunk (3 aligned float4 loads), so every
// element's w-phase is a compile-time constant -> accumulators stay in VGPRs.
// grid = (168, 16), block = 256  ->  168 * 4096 chunks, exact cover, no tail.
// ---------------------------------------------------------------------------
__global__ __launch_bounds__(256) void rms_partials_kernel(
    const float* __restrict__ x, const float* __restrict__ y,
    const unsigned char* __restrict__ ok, float* __restrict__ ws) {
  const int g = blockIdx.x;                                // (b*F + f)
  const int c = blockIdx.y * 256 + threadIdx.x;            // chunk in [0,4096)
  const int base4 = g * (P_CNT * W_CNT / 4) + c * 3;       // float4 index

  const float4* __restrict__ X4 = (const float4*)x;
  const float4* __restrict__ Y4 = (const float4*)y;
  const uchar4* __restrict__ O4 = (const uchar4*)ok;

  float4 xv[3], yv[3];
  uchar4 ov[3];
#pragma unroll
  for (int j = 0; j < 3; ++j) {
    xv[j] = X4[base4 + j];
    yv[j] = Y4[base4 + j];
    ov[j] = O4[base4 + j];
  }

  float an[3]  = {0.f, 0.f, 0.f};
  float asx[3] = {0.f, 0.f, 0.f};
  float asy[3] = {0.f, 0.f, 0.f};
  float as2[3] = {0.f, 0.f, 0.f};
#pragma unroll
  for (int j = 0; j < 3; ++j) {
    const float xs[4] = {xv[j].x, xv[j].y, xv[j].z, xv[j].w};
    const float ys[4] = {yv[j].x, yv[j].y, yv[j].z, yv[j].w};
    const unsigned char os[4] = {ov[j].x, ov[j].y, ov[j].z, ov[j].w};
#pragma unroll
    for (int k = 0; k < 4; ++k) {
      const int w = (j * 4 + k) % 3;                       // compile-time const
      const float m = os[k] ? 1.0f : 0.0f;
      an[w]  += m;
      asx[w] += xs[k] * m;
      asy[w] += ys[k] * m;
      as2[w] += (xs[k] * xs[k] + ys[k] * ys[k]) * m;
    }
  }

  __shared__ float red[12];
  if (threadIdx.x < 12) red[threadIdx.x] = 0.0f;
  __syncthreads();
#pragma unroll
  for (int w = 0; w < 3; ++w) {
    atomicAdd(&red[w * 4 + 0], an[w]);                     // ds_add_f32
    atomicAdd(&red[w * 4 + 1], asx[w]);
    atomicAdd(&red[w * 4 + 2], asy[w]);
    atomicAdd(&red[w * 4 + 3], as2[w]);
  }
  __syncthreads();
  if (threadIdx.x < 12)
    atomicAdd(&ws[g * 12 + threadIdx.x], red[threadIdx.x]);
}

// ---------------------------------------------------------------------------
// Kernel 2: sumQ reduction over the three (S,B,F,P,W) penalty stacks using
// V_WMMA_F32_16X16X4_F32 with B = all-ones as an fp32 row-sum engine.
// D[m][n] += sum_k A[m][k], so Sum(D over all 8 acc VGPRs x 32 lanes) = 16 * sumA.
// Exactly 8 scalar-uniform iterations (1008 blocks * 256 thr * 8 * float4
// == 8,257,536 elements), so EXEC is all-ones at every WMMA as required.
// ---------------------------------------------------------------------------
__global__ __launch_bounds__(256) void q_reduce_wmma_kernel(
    const float* __restrict__ th, const float* __restrict__ tp,
    const float* __restrict__ zr, float* __restrict__ ws) {
  const long long N4 = NELEM / 4;                          // 2,064,384
  const long long stride = 1008LL * 256LL;                 // 258,048
  const long long tid = (long long)blockIdx.x * 256 + threadIdx.x;

  const float4* __restrict__ T = (const float4*)th;
  const float4* __restrict__ U = (const float4*)tp;
  const float4* __restrict__ Z = (const float4*)zr;

  v8f acc = {};
  const v2f ones = {1.0f, 1.0f};

  for (int it = 0; it < 8; ++it) {                         // scalar-uniform trip
    const long long i = tid + (long long)it * stride;
    const float4 a0 = T[i], a1 = T[i + N4], a2 = T[i + 2 * N4];
    const float4 b0 = U[i], b1 = U[i + N4], b2 = U[i + 2 * N4];
    const float4 c0 = Z[i], c1 = Z[i + N4], c2 = Z[i + 2 * N4];

    float4 q;
    q.x = a0.x + a1.x + a2.x + b0.x + b1.x + b2.x + c0.x + c1.x + c2.x;
    q.y = a0.y + a1.y + a2.y + b0.y + b1.y + b2.y + c0.y + c1.y + c2.y;
    q.z = a0.z + a1.z + a2.z + b0.z + b1.z + b2.z + c0.z + c1.z + c2.z;
    q.w = a0.w + a1.w + a2.w + b0.w + b1.w + b2.w + c0.w + c1.w + c2.w;
    // NaN -> 0 (matches jnp.where(isnan(Q), 0, Q))
    q.x = (q.x == q.x) ? q.x : 0.0f;
    q.y = (q.y == q.y) ? q.y : 0.0f;
    q.z = (q.z == q.z) ? q.z : 0.0f;
    q.w = (q.w == q.w) ? q.w : 0.0f;

    const v2f A0 = {q.x, q.y};
    const v2f A1 = {q.z, q.w};
    // (neg_a, A, neg_b, B, c_mod, C, reuse_a, reuse_b)
    acc = __builtin_amdgcn_wmma_f32_16x16x4_f32(false, A0, false, ones,
                                                (short)0, acc, false, false);
    acc = __builtin_amdgcn_wmma_f32_16x16x4_f32(false, A1, false, ones,
                                                (short)0, acc, false, false);
  }

  float p = acc[0] + acc[1] + acc[2] + acc[3] +
            acc[4] + acc[5] + acc[6] + acc[7];

  __shared__ float sblk;
  if (threadIdx.x == 0) sblk = 0.0f;
  __syncthreads();
  atomicAdd(&sblk, p);                                     // ds_add_f32
  __syncthreads();
  if (threadIdx.x == 0) atomicAdd(&ws[WS_SUMQ], sblk);     // global_atomic_add_f32
}

// ---------------------------------------------------------------------------
// Kernel 3: finalize. 504 (b,f,w) groups -> rms, combine with sumQ.
// ---------------------------------------------------------------------------
__global__ __launch_bounds__(512) void finalize_kernel(
    const float* __restrict__ ws, float* __restrict__ out) {
  __shared__ float red[512];
  const int t = threadIdx.x;
  float r = 0.0f;
  if (t < GROUPS * W_CNT) {                                // 504
    const int g = t / 3, w = t % 3;
    float n  = ws[g * 12 + w * 4 + 0];
    float sx = ws[g * 12 + w * 4 + 1];
    float sy = ws[g * 12 + w * 4 + 2];
    float s2 = ws[g * 12 + w * 4 + 3];
    n = fmaxf(n, 1.0f);
    // sum((x-xc)^2 + (y-yc)^2)*ok = s2 - (sx^2 + sy^2)/n ; msr = ... / n
    const float msr = (s2 - (sx * sx + sy * sy) / n) / n;
    r = sqrtf(msr + 1e-12f);
  }
  red[t] = r;
  __syncthreads();
  for (int off = 256; off > 0; off >>= 1) {
    if (t < off) red[t] += red[t + off];
    __syncthreads();
  }
  if (t == 0) {
    const float rms  = red[0] / (float)(GROUPS * W_CNT);
    const float sumQ = ws[WS_SUMQ] / (16.0f * (float)S_CNT); // /16 col-replication, /S
    out[0] = rms + 0.2f * sumQ;
  }
}

extern "C" void kernel_launch(void* const* d_in, const int* in_sizes, int n_in,
                              void* d_out, int out_size, void* d_ws, size_t ws_size,
                              hipStream_t stream) {
  (void)in_sizes; (void)n_in; (void)out_size; (void)ws_size;
  const float* x  = (const float*)d_in[0];
  const float* y  = (const float*)d_in[1];
  const float* th = (const float*)d_in[2];
  const float* tp = (const float*)d_in[3];
  const float* zr = (const float*)d_in[4];
  const unsigned char* ok = (const unsigned char*)d_in[5]; // bool array, 1 byte/elt
  float* out = (float*)d_out;
  float* ws  = (float*)d_ws;

  hipLaunchKernelGGL(zero_ws_kernel, dim3((WS_FLOATS + 255) / 256), dim3(256),
                     0, stream, ws);
  hipLaunchKernelGGL(rms_partials_kernel, dim3(GROUPS, 16), dim3(256),
                     0, stream, x, y, ok, ws);
  hipLaunchKernelGGL(q_reduce_wmma_kernel, dim3(1008), dim3(256),
                     0, stream, th, tp, zr, ws);
  hipLaunchKernelGGL(finalize_kernel, dim3(1), dim3(512),
                     0, stream, ws, out);
}